// StaticGNN_73512660238836
// MI455X (gfx1250) — compile-verified
//
#include <hip/hip_runtime.h>
#include <hip/hip_bf16.h>

typedef __attribute__((ext_vector_type(2))) float v2f;
typedef __attribute__((ext_vector_type(8))) float v8f;

#define D 128

// ---------------------------------------------------------------------------
// Degree / normalization
// ---------------------------------------------------------------------------
__global__ void zero_f32_kernel(float* __restrict__ p, int n) {
    int i = blockIdx.x * blockDim.x + threadIdx.x;
    if (i < n) p[i] = 0.0f;
}

__global__ void deg_count_kernel(const int* __restrict__ dst, float* __restrict__ deg, int E) {
    int i = blockIdx.x * blockDim.x + threadIdx.x;
    if (i < E) atomicAdd(&deg[dst[i]], 1.0f);
}

__global__ void make_dinv_kernel(const float* __restrict__ deg, float* __restrict__ dinv, int N) {
    int i = blockIdx.x * blockDim.x + threadIdx.x;
    if (i < N) dinv[i] = rsqrtf(deg[i] + 1.0f);
}

// ---------------------------------------------------------------------------
// Fused WMMA GEMM + self-loop/bias epilogue:
//   hw  = (RELU? relu(H) : H) @ W          [rows,128]
//   agg = hw * dinv[row]^2 + bias[col]     (seed for edge scatter)
//
// Block: 256 thr = 8 waves, computes 32 rows x 128 cols.
// Each wave owns one 16-col slab and TWO 16x16 row tiles -> each B fragment
// (2 ds loads) feeds 2 x V_WMMA_F32_16X16X4_F32; K=128 in steps of 4.
// A frag (16x4 f32): lanes 0-15 -> M=lane, {K=kk,kk+1}; lanes 16-31 -> {kk+2,kk+3}
// B frag (4x16 f32): lanes 0-15 -> N=lane, {K=kk,kk+1}; lanes 16-31 -> {kk+2,kk+3}
// C/D (16x16 f32):  VGPR j: lanes 0-15 -> M=j; lanes 16-31 -> M=j+8; N=lane&15
// ---------------------------------------------------------------------------
template <int RELU>
__global__ __launch_bounds__(256) void gcn_gemm_wmma_kernel(
    const float* __restrict__ H, const float* __restrict__ W,
    const float* __restrict__ dinv, const float* __restrict__ bias,
    float* __restrict__ HW, float* __restrict__ AGG, int nrows)
{
    __shared__ float sW[D * D];   // 64 KB: full weight matrix in LDS

    const int tid = threadIdx.x;
    for (int i = tid; i < (D * D) / 4; i += 256) {
        ((float4*)sW)[i] = ((const float4*)W)[i];
    }
    __syncthreads();

    const int wave    = tid >> 5;        // 0..7 -> column slab
    const int lane    = tid & 31;
    const int laneM   = lane & 15;
    const int hiHalf  = lane >> 4;       // 0: lanes 0-15, 1: lanes 16-31
    const int rowBase = blockIdx.x * 32;
    const int colBase = wave * 16;
    if (rowBase >= nrows) return;

    const int r0 = rowBase + laneM;                               // tile 0 row
    const int r1 = min(rowBase + 16 + laneM, nrows - 1);          // tile 1 row (clamped)
    const float* __restrict__ hrow0 = H + (size_t)r0 * D;
    const float* __restrict__ hrow1 = H + (size_t)r1 * D;
    const float* __restrict__ wcol  = sW + colBase + laneM;

    v8f acc0 = {};
    v8f acc1 = {};
    #pragma unroll 4
    for (int kk = 0; kk < D; kk += 4) {
        const int ka = kk + hiHalf * 2;
        float a00 = hrow0[ka];
        float a01 = hrow0[ka + 1];
        float a10 = hrow1[ka];
        float a11 = hrow1[ka + 1];
        if (RELU) {
            a00 = fmaxf(a00, 0.0f); a01 = fmaxf(a01, 0.0f);
            a10 = fmaxf(a10, 0.0f); a11 = fmaxf(a11, 0.0f);
        }
        v2f af0 = { a00, a01 };
        v2f af1 = { a10, a11 };
        v2f bf  = { wcol[(size_t)ka * D], wcol[(size_t)(ka + 1) * D] };
        acc0 = __builtin_amdgcn_wmma_f32_16x16x4_f32(
            false, af0, false, bf, (short)0, acc0, false, false);
        acc1 = __builtin_amdgcn_wmma_f32_16x16x4_f32(
            false, af1, false, bf, (short)0, acc1, false, false);
    }

    // Epilogue: write hw and seed agg = hw*dinv^2 + bias
    const int mOff = hiHalf * 8;
    const int col  = colBase + laneM;
    const float bcol = bias[col];

    #pragma unroll
    for (int j = 0; j < 8; ++j) {
        int r = rowBase + mOff + j;
        float v = acc0[j];
        float di = dinv[r];
        HW [(size_t)r * D + col] = v;
        AGG[(size_t)r * D + col] = v * di * di + bcol;
    }
    #pragma unroll
    for (int j = 0; j < 8; ++j) {
        int r = rowBase + 16 + mOff + j;
        if (r < nrows) {
            float v = acc1[j];
            float di = dinv[r];
            HW [(size_t)r * D + col] = v;
            AGG[(size_t)r * D + col] = v * di * di + bcol;
        }
    }
}

// ---------------------------------------------------------------------------
// Edge scatter: agg[dst] += hw[src] * (dinv[src]*dinv[dst])
// 32 lanes per edge, float4 gather, 4 global f32 atomics per lane (L2-resident).
// ---------------------------------------------------------------------------
__global__ __launch_bounds__(256) void scatter_edges_kernel(
    const float* __restrict__ hw, const int* __restrict__ src,
    const int* __restrict__ dst, const float* __restrict__ dinv,
    float* __restrict__ agg, int E)
{
    long long t = (long long)blockIdx.x * 256 + threadIdx.x;
    long long e = t >> 5;
    if (e >= E) return;
    int lane = (int)(t & 31);
    int s = src[e];
    int d = dst[e];
    float norm = dinv[s] * dinv[d];
    float4 v = ((const float4*)(hw + (size_t)s * D))[lane];
    float* ap = agg + (size_t)d * D + lane * 4;
    atomicAdd(ap + 0, v.x * norm);
    atomicAdd(ap + 1, v.y * norm);
    atomicAdd(ap + 2, v.z * norm);
    atomicAdd(ap + 3, v.w * norm);
}

// ---------------------------------------------------------------------------
// Head: out[n] = dot(relu(h[n]), head_W) + head_b   (wave32 reduction)
// ---------------------------------------------------------------------------
__global__ __launch_bounds__(256) void head_kernel(
    const float* __restrict__ h, const float* __restrict__ headW,
    const float* __restrict__ headb, float* __restrict__ out, int N)
{
    int t = blockIdx.x * 256 + threadIdx.x;
    int n = t >> 5;
    if (n >= N) return;
    int lane = t & 31;
    float4 v = ((const float4*)(h + (size_t)n * D))[lane];
    float4 w = ((const float4*)headW)[lane];
    float s = fmaxf(v.x, 0.0f) * w.x + fmaxf(v.y, 0.0f) * w.y +
              fmaxf(v.z, 0.0f) * w.z + fmaxf(v.w, 0.0f) * w.w;
    #pragma unroll
    for (int off = 16; off > 0; off >>= 1)
        s += __shfl_xor(s, off, 32);
    if (lane == 0) out[n] = s + headb[0];
}

// ---------------------------------------------------------------------------
// Launch
// ---------------------------------------------------------------------------
extern "C" void kernel_launch(void* const* d_in, const int* in_sizes, int n_in,
                              void* d_out, int out_size, void* d_ws, size_t ws_size,
                              hipStream_t stream) {
    const float* x     = (const float*)d_in[0];   // [N,128]
    const int*   eidx  = (const int*)d_in[1];     // [2,E]
    const float* Ws    = (const float*)d_in[2];   // [L,128,128]
    const float* bs    = (const float*)d_in[3];   // [L,128]
    const float* headW = (const float*)d_in[4];   // [128,1]
    const float* headb = (const float*)d_in[5];   // [1]

    const int N = in_sizes[0] / D;
    const int E = in_sizes[1] / 2;
    const int L = in_sizes[2] / (D * D);

    const int* src = eidx;
    const int* dst = eidx + E;

    // workspace layout
    float* deg  = (float*)d_ws;                  // N
    float* dinv = deg + N;                       // N
    float* buf0 = dinv + N;                      // N*128  (hw, reused every layer)
    float* buf1 = buf0 + (size_t)N * D;          // N*128  (agg, alternating)
    float* buf2 = buf1 + (size_t)N * D;          // N*128  (agg, alternating)

    const int T = 256;

    // 1) degrees + symmetric-normalization factors
    zero_f32_kernel<<<(N + T - 1) / T, T, 0, stream>>>(deg, N);
    deg_count_kernel<<<(E + T - 1) / T, T, 0, stream>>>(dst, deg, E);
    make_dinv_kernel<<<(N + T - 1) / T, T, 0, stream>>>(deg, dinv, N);

    const int gemmGrid = (N + 31) / 32;
    const int nodeGrid = (int)(((long long)N * 32 + T - 1) / T);
    const int edgeGrid = (int)(((long long)E * 32 + T - 1) / T);

    // 2) GCN layers: fused GEMM+self/bias epilogue, then edge scatter.
    //    hw always in buf0 (free after each scatter); agg alternates buf1/buf2.
    const float* hin = x;
    for (int l = 0; l < L; ++l) {
        float* hw  = buf0;
        float* agg = (l & 1) ? buf2 : buf1;
        if (l == 0) {
            gcn_gemm_wmma_kernel<0><<<gemmGrid, T, 0, stream>>>(
                hin, Ws + (size_t)l * D * D, dinv, bs + (size_t)l * D, hw, agg, N);
        } else {
            gcn_gemm_wmma_kernel<1><<<gemmGrid, T, 0, stream>>>(
                hin, Ws + (size_t)l * D * D, dinv, bs + (size_t)l * D, hw, agg, N);
        }
        scatter_edges_kernel<<<edgeGrid, T, 0, stream>>>(hw, src, dst, dinv, agg, E);
        hin = agg;
    }

    // 3) head (ReLU fused)
    head_kernel<<<nodeGrid, T, 0, stream>>>(hin, headW, headb, (float*)d_out, N);
}